// TimeLSTMCell_3_64089501990951
// MI455X (gfx1250) — compile-verified
//
#include <hip/hip_runtime.h>

typedef float v2f __attribute__((ext_vector_type(2)));
typedef float v4f __attribute__((ext_vector_type(4)));
typedef float v8f __attribute__((ext_vector_type(8)));

#define BB 4096
#define DD 256
#define UU 512

constexpr int BM  = 64;    // rows per block
constexpr int BN  = 64;    // u-columns per block
constexpr int KC  = 32;    // K chunk
constexpr int AS  = 36;    // LDS dword stride for A tile rows (conflict-free, 16B aligned)
constexpr int PR  = 160;   // LDS dword stride per K-pair row of W (== 32 mod 64 -> conflict-free b64 loads)
constexpr int GS  = (KC / 2) * PR;   // dwords per gate in sW = 16*160 = 2560

__device__ __forceinline__ float sigf(float v) { return 1.0f / (1.0f + __expf(-v)); }
__device__ __forceinline__ float tanhfast(float v) { return 2.0f * sigf(2.0f * v) - 1.0f; }

// 32-bit LDS byte offset of a generic pointer to __shared__ (low 32 bits of flat addr)
__device__ __forceinline__ unsigned lds_addr32(const void* p) {
    return (unsigned)(unsigned long long)p;
}

// CDNA5 async copy: global -> LDS, 16B per lane, tracked by ASYNCcnt
__device__ __forceinline__ void async_copy_b128(unsigned lds_off, const float* gaddr) {
    asm volatile("global_load_async_to_lds_b128 %0, %1, off"
                 :: "v"(lds_off), "v"(gaddr) : "memory");
}
__device__ __forceinline__ void wait_asynccnt0() {
    asm volatile("s_wait_asynccnt 0x0" ::: "memory");
}

__global__ __launch_bounds__(256)
void time_lstm_fused(const float* __restrict__ x,  const float* __restrict__ t,
                     const float* __restrict__ h0, const float* __restrict__ c0,
                     const float* __restrict__ wk, const float* __restrict__ rk,
                     const float* __restrict__ kt, float* __restrict__ out)
{
    __shared__ __align__(16) float sA[BM * AS];       //  9216 B
    __shared__ __align__(16) float sW[5 * GS];        // 51200 B (K-pair interleaved)
    __shared__ float sT[BM];
    __shared__ float sKT[3 * BN];

    const int tid  = threadIdx.x;
    const int wave = tid >> 5;
    const int lane = tid & 31;
    const int half = lane >> 4;            // 0: lanes 0-15, 1: lanes 16-31
    const int lm   = lane & 15;
    const int m_off = (wave & 3) * 16;     // wave's 16-row block
    const int n_off = (wave >> 2) * 32;    // wave's 32-col block

    const int u0 = blockIdx.x * BN;
    const int m0 = blockIdx.y * BM;

    if (tid < BM)  sT[tid] = t[m0 + tid];
    if (tid < 3 * BN) {
        int g = tid / BN, c = tid % BN;
        sKT[tid] = kt[g * UU + u0 + c];
    }

    v8f acc[8][2];
    #pragma unroll
    for (int g = 0; g < 8; ++g) {
        #pragma unroll
        for (int ct = 0; ct < 2; ++ct) {
            v8f z = {0.f, 0.f, 0.f, 0.f, 0.f, 0.f, 0.f, 0.f};
            acc[g][ct] = z;
        }
    }

    // B-fragment LDS base for this wave/lane: pair-interleaved layout
    // element (gate g, K=k, col n) lives at  g*GS + (k>>1)*PR + n*2 + (k&1)
    const int bcol0 = (n_off + lm) * 2;          // ct=0 tile
    const int bcol1 = (n_off + 16 + lm) * 2;     // ct=1 tile
    const int arow  = (m_off + lm) * AS;

    // ---------------- Phase 1: X (64xD) @ kernel -> gates [i,c,o,t1,t2] ----------------
    for (int k0 = 0; k0 < DD; k0 += KC) {
        __syncthreads();
        // A tile 64x32 via async-to-LDS (2 x b128 per thread)
        #pragma unroll
        for (int it = 0; it < 2; ++it) {
            int j = tid + it * 256;
            int row = j >> 3, c4 = (j & 7) << 2;
            async_copy_b128(lds_addr32(&sA[row * AS + c4]),
                            &x[(m0 + row) * DD + k0 + c4]);
        }
        // W tile: 5 gates x (16 K-pairs x 64 cols), register-interleaved (5 units/thread)
        #pragma unroll
        for (int it = 0; it < 5; ++it) {
            int uj = tid + it * 256;
            int g = uj >> 8, rem = uj & 255;
            int p = rem >> 4, c = (rem & 15) << 2;
            const float* src = &wk[(k0 + 2 * p) * (5 * UU) + g * UU + u0 + c];
            v4f lo = *(const v4f*)src;
            v4f hi = *(const v4f*)(src + 5 * UU);
            float* dst = &sW[g * GS + p * PR + 2 * c];
            v4f p0 = {lo.x, hi.x, lo.y, hi.y};
            v4f p1 = {lo.z, hi.z, lo.w, hi.w};
            *(v4f*)dst       = p0;
            *(v4f*)(dst + 4) = p1;
        }
        wait_asynccnt0();
        __syncthreads();

        #pragma unroll
        for (int kk = 0; kk < KC; kk += 4) {
            v2f a = *(const v2f*)&sA[arow + kk + 2 * half];          // K=kk+2h, kk+2h+1
            const int prow = (kk >> 1) + half;                       // K-pair row
            #pragma unroll
            for (int g = 0; g < 5; ++g) {
                const float* gb = &sW[g * GS + prow * PR];
                v2f b0 = *(const v2f*)(gb + bcol0);
                v2f b1 = *(const v2f*)(gb + bcol1);
                acc[g][0] = __builtin_amdgcn_wmma_f32_16x16x4_f32(
                    false, a, false, b0, (short)0, acc[g][0], false, false);
                acc[g][1] = __builtin_amdgcn_wmma_f32_16x16x4_f32(
                    false, a, false, b1, (short)0, acc[g][1], false, false);
            }
        }
    }

    // ---------------- Phase 2: H0 (64xU) @ recurrent -> gates [i,c,o] ----------------
    for (int k0 = 0; k0 < UU; k0 += KC) {
        __syncthreads();
        #pragma unroll
        for (int it = 0; it < 2; ++it) {
            int j = tid + it * 256;
            int row = j >> 3, c4 = (j & 7) << 2;
            async_copy_b128(lds_addr32(&sA[row * AS + c4]),
                            &h0[(m0 + row) * UU + k0 + c4]);
        }
        #pragma unroll
        for (int it = 0; it < 3; ++it) {
            int uj = tid + it * 256;
            int g = uj >> 8, rem = uj & 255;
            int p = rem >> 4, c = (rem & 15) << 2;
            const float* src = &rk[(k0 + 2 * p) * (3 * UU) + g * UU + u0 + c];
            v4f lo = *(const v4f*)src;
            v4f hi = *(const v4f*)(src + 3 * UU);
            float* dst = &sW[g * GS + p * PR + 2 * c];
            v4f p0 = {lo.x, hi.x, lo.y, hi.y};
            v4f p1 = {lo.z, hi.z, lo.w, hi.w};
            *(v4f*)dst       = p0;
            *(v4f*)(dst + 4) = p1;
        }
        wait_asynccnt0();
        __syncthreads();

        #pragma unroll
        for (int kk = 0; kk < KC; kk += 4) {
            v2f a = *(const v2f*)&sA[arow + kk + 2 * half];
            const int prow = (kk >> 1) + half;
            #pragma unroll
            for (int g = 0; g < 3; ++g) {
                const float* gb = &sW[g * GS + prow * PR];
                v2f b0 = *(const v2f*)(gb + bcol0);
                v2f b1 = *(const v2f*)(gb + bcol1);
                acc[5 + g][0] = __builtin_amdgcn_wmma_f32_16x16x4_f32(
                    false, a, false, b0, (short)0, acc[5 + g][0], false, false);
                acc[5 + g][1] = __builtin_amdgcn_wmma_f32_16x16x4_f32(
                    false, a, false, b1, (short)0, acc[5 + g][1], false, false);
            }
        }
    }

    // ---------------- Epilogue: fused TimeLSTM math, pure per-lane VALU ----------------
    const float NEG_EPS = -1e-5f;
    #pragma unroll
    for (int ct = 0; ct < 2; ++ct) {
        const int ucl = n_off + ct * 16 + lm;
        const int gu  = u0 + ucl;
        const float ktv1 = sKT[ucl];
        const float ktv2 = sKT[BN + ucl];
        const float ktvo = sKT[2 * BN + ucl];
        #pragma unroll
        for (int i = 0; i < 8; ++i) {
            const int lrow = m_off + i + 8 * half;    // C/D layout: VGPR i -> M=i / M=8+i
            const int grow = m0 + lrow;
            const float tval = sT[lrow];
            const float c0v  = c0[grow * UU + gu];

            const float x_i  = acc[0][ct][i];
            const float x_c  = acc[1][ct][i];
            const float x_o  = acc[2][ct][i];
            const float x_t1 = acc[3][ct][i];
            const float x_t2 = acc[4][ct][i];
            const float r_i  = acc[5][ct][i];
            const float r_c  = acc[6][ct][i];
            const float r_o  = acc[7][ct][i];

            const float ig  = sigf(x_i + r_i);
            const float t1  = sigf(x_t1 + sigf(tval * ktv1));
            const float t1c = (t1 > NEG_EPS) ? NEG_EPS : t1;
            const float t2  = sigf(x_t2 + sigf(tval * ktv2));
            const float ctl = tanhfast(x_c + r_c);
            const float cm_ = (1.0f - ig * t1) * c0v + ig * ctl * t1c;
            const float cm  = (1.0f - ig) * c0v + ig * ctl * t2;
            const float o   = sigf(x_o + r_o + tval * ktvo);
            const float h   = tanhfast(cm_) * o;

            __builtin_nontemporal_store(h,  &out[grow * UU + gu]);            // h
            __builtin_nontemporal_store(cm, &out[BB * UU + grow * UU + gu]);  // c_m
        }
    }
}

extern "C" void kernel_launch(void* const* d_in, const int* in_sizes, int n_in,
                              void* d_out, int out_size, void* d_ws, size_t ws_size,
                              hipStream_t stream) {
    const float* x  = (const float*)d_in[0];
    const float* t  = (const float*)d_in[1];
    const float* h0 = (const float*)d_in[2];
    const float* c0 = (const float*)d_in[3];
    const float* wk = (const float*)d_in[4];
    const float* rk = (const float*)d_in[5];
    const float* kt = (const float*)d_in[6];
    float* out = (float*)d_out;

    dim3 grid(UU / BN, BB / BM);   // 8 x 64
    dim3 block(256);
    time_lstm_fused<<<grid, block, 0, stream>>>(x, t, h0, c0, wk, rk, kt, out);
}